// QuantizedWeight_41583873359892
// MI455X (gfx1250) — compile-verified
//
#include <hip/hip_runtime.h>
#include <hip/hip_bf16.h>
#include <stdint.h>

// AQLM 2x16 dequantization: out[o, i*8..i*8+7] = cb0[codes[o,i,0]] + cb1[codes[o,i,1]]
#define NUM_CB        2
#define CB_SIZE       65536
#define IN_GROUP      8
#define NOG           4096
#define NIG           1376
#define IN_FEATURES   (NIG * IN_GROUP)    // 11008
#define CODES_PER_ROW (NIG * NUM_CB)      // 2752 int32 per out-row
#define TO            2                   // out rows per workgroup
#define THREADS       256                 // 8 wave32s

typedef unsigned int v4u __attribute__((ext_vector_type(4)));
typedef int          v8i __attribute__((ext_vector_type(8)));
typedef int          v4i __attribute__((ext_vector_type(4)));
typedef float        v4f __attribute__((ext_vector_type(4)));  // native vector: OK for NT builtins

__global__ __launch_bounds__(THREADS) void aqlm_dequant_kernel(
    const int*   __restrict__ codes,
    const float* __restrict__ codebooks,
    float*       __restrict__ out)
{
    __shared__ __align__(16) int lds_codes[TO * CODES_PER_ROW]; // 22016 B

    const int row_base = blockIdx.x * TO;

    // ---- TDM: DMA this block's codes tile (TO x 2752 int32) into LDS ----
    // Issued by wave 0 only (one tensor_load_to_lds per workgroup).
    if (threadIdx.x < 32) {
        const uint64_t gaddr = (uint64_t)(uintptr_t)(codes + (size_t)row_base * CODES_PER_ROW);
        const uint32_t laddr = (uint32_t)(uintptr_t)(&lds_codes[0]); // low 32 bits = LDS offset

        // D# group0: count=1(valid), lds_addr, global_addr[56:0], type=2
        v4u g0;
        g0[0] = 1u;
        g0[1] = laddr;
        g0[2] = (uint32_t)gaddr;
        g0[3] = (uint32_t)(gaddr >> 32) | (2u << 30);

        // D# group1: mask=0, data_size=2(4B), no barrier/iterate/pad,
        // tensor_dim0=2752, tensor_dim1=4096, tile_dim0=2752, tile_dim1=TO,
        // tile_dim2=0, tensor_dim0_stride=2752, tensor_dim1_stride=0 (2D tile)
        v8i g1;
        g1[0] = (2 << 16);
        g1[1] = (int)((CODES_PER_ROW & 0xFFFF) << 16);
        g1[2] = (int)(((unsigned)CODES_PER_ROW >> 16) | ((NOG & 0xFFFF) << 16));
        g1[3] = (int)(((unsigned)NOG >> 16) | ((unsigned)CODES_PER_ROW << 16));
        g1[4] = TO;           // tile_dim1=TO, tile_dim2=0
        g1[5] = CODES_PER_ROW;
        g1[6] = 0;
        g1[7] = 0;

        v4i g2 = {0, 0, 0, 0};   // tensor_dim2/3 unused (tile_dim3=0)
        v4i g3 = {0, 0, 0, 0};   // tensor_dim4 unused (tile_dim4=0)

#if defined(__clang_major__) && (__clang_major__ >= 23)
        v8i g4 = {0, 0, 0, 0, 0, 0, 0, 0};
        __builtin_amdgcn_tensor_load_to_lds(g0, g1, g2, g3, g4, 0);
#else
        __builtin_amdgcn_tensor_load_to_lds(g0, g1, g2, g3, 0);
#endif
        __builtin_amdgcn_s_wait_tensorcnt(0);
    }
    __syncthreads();

    // ---- Gather from L2-resident codebooks, add, NT-store wide ----
    const float* __restrict__ cb0 = codebooks;
    const float* __restrict__ cb1 = codebooks + (size_t)CB_SIZE * IN_GROUP;

    for (int row = 0; row < TO; ++row) {
        float* outrow = out + (size_t)(row_base + row) * IN_FEATURES;
        const int* crow = &lds_codes[row * CODES_PER_ROW];
        for (int i = threadIdx.x; i < NIG; i += THREADS) {
            const int c0 = crow[2 * i];       // ds_load_b64 (consecutive pair)
            const int c1 = crow[2 * i + 1];

            const v4f* p0 = (const v4f*)(cb0 + (size_t)(uint32_t)c0 * IN_GROUP);
            const v4f* p1 = (const v4f*)(cb1 + (size_t)(uint32_t)c1 * IN_GROUP);
            const v4f a0 = p0[0];             // global_load_b128 gathers (L2-hot)
            const v4f a1 = p0[1];
            const v4f b0 = p1[0];
            const v4f b1 = p1[1];

            const v4f r0 = a0 + b0;
            const v4f r1 = a1 + b1;

            v4f* dst = (v4f*)(outrow + (size_t)i * IN_GROUP);
            __builtin_nontemporal_store(r0, dst);      // keep 180MB stream out of L2
            __builtin_nontemporal_store(r1, dst + 1);  // (protect 4MB hot codebook)
        }
    }
}

extern "C" void kernel_launch(void* const* d_in, const int* in_sizes, int n_in,
                              void* d_out, int out_size, void* d_ws, size_t ws_size,
                              hipStream_t stream) {
    const int*   codes     = (const int*)d_in[0];     // [4096, 1376, 2] int32
    const float* codebooks = (const float*)d_in[1];   // [2, 65536, 1, 8] float32
    float*       out       = (float*)d_out;           // [4096, 11008] float32

    dim3 grid(NOG / TO);  // 2048 workgroups
    aqlm_dequant_kernel<<<grid, THREADS, 0, stream>>>(codes, codebooks, out);
}